// KarplusStrongDecoder_154618823275
// MI455X (gfx1250) — compile-verified
//
#include <hip/hip_runtime.h>
#include <stdint.h>

// Karplus-Strong decoder for MI455X (gfx1250).
// 128 voices -> 4 single-wave (32-lane) blocks, one lane per voice.
// Delay lines live in LDS (bank-conflict-free [k*32+lane] layout).
// Noise input is DMA'd in with TENSOR_LOAD_TO_LDS; output tiles are staged
// in LDS and flushed with TENSOR_STORE_FROM_LDS, double-buffered via TENSORcnt.

#define SR_I   16000
#define MAXD   134      // reference MAX_DELAY (buffer width; JAX clamps/drops OOB)
#define NVOICE 128
#define VPB    32       // voices per block == one wave32
#define CHUNK  128      // time steps per staged output tile

typedef unsigned int u32x4 __attribute__((ext_vector_type(4)));
typedef int          i32x4 __attribute__((ext_vector_type(4)));
typedef int          i32x8 __attribute__((ext_vector_type(8)));

// Low 32 bits of a flat pointer into shared memory == LDS byte address
// (flat LDS aperture keeps the offset in addr[31:0], ISA §10.2).
__device__ __forceinline__ uint32_t lds_byte_addr(const void* p) {
  return (uint32_t)(uintptr_t)p;
}

// Build the D# for a compact LDS tile of tileW x tileH f32 elements mapping
// to a 2D global region with row stride rowStrideElems (elements).
// Layout per CDNA5 ISA.
__device__ __forceinline__ void tdm_make_desc(u32x4& g0, i32x8& g1,
                                              uint32_t lds_addr,
                                              const float* gptr,
                                              int tileW, int tileH,
                                              int rowStrideElems) {
  uint64_t ga = (uint64_t)(uintptr_t)gptr;
  g0.x = 1u;                                   // count=1, is_restore=0, gather off
  g0.y = lds_addr;                             // lds_addr (bytes)
  g0.z = (uint32_t)ga;                         // global_addr[31:0]
  g0.w = (uint32_t)((ga >> 32) & 0x01FFFFFFu)  // global_addr[56:32]
       | (2u << 30);                           // type = 2 ("image")
  uint32_t d0 = (uint32_t)tileW;               // tensor_dim0 (elements)
  uint32_t d1 = (uint32_t)tileH;               // tensor_dim1 (rows)
  uint32_t s0 = (uint32_t)rowStrideElems;      // tensor_dim0_stride
  g1[0] = (int)(2u << 16);                                   // data_size=2 (4B), mask=0
  g1[1] = (int)((d0 & 0xFFFFu) << 16);                       // tensor_dim0[15:0] @bit48
  g1[2] = (int)((d0 >> 16) | ((d1 & 0xFFFFu) << 16));        // dim0 hi | dim1 lo
  g1[3] = (int)((d1 >> 16) | ((uint32_t)tileW << 16));       // dim1 hi | tile_dim0
  g1[4] = (int)((uint32_t)tileH & 0xFFFFu);                  // tile_dim1, tile_dim2=0
  g1[5] = (int)s0;                                           // dim0_stride[31:0]
  g1[6] = 0;                                                 // dim0_stride hi, dim1_stride lo
  g1[7] = 0;
}

__device__ __forceinline__ void tdm_store_tile(uint32_t lds_addr, const float* gptr,
                                               int tileW, int tileH, int rowStrideElems) {
  u32x4 g0; i32x8 g1;
  tdm_make_desc(g0, g1, lds_addr, gptr, tileW, tileH, rowStrideElems);
  i32x4 gz = {0, 0, 0, 0};
#if defined(__clang_major__) && (__clang_major__ >= 23)
  i32x8 gz8 = {0, 0, 0, 0, 0, 0, 0, 0};
  __builtin_amdgcn_tensor_store_from_lds(g0, g1, gz, gz, gz8, 0);
#else
  __builtin_amdgcn_tensor_store_from_lds(g0, g1, gz, gz, 0);
#endif
}

__device__ __forceinline__ void tdm_load_tile(uint32_t lds_addr, const float* gptr,
                                              int tileW, int tileH, int rowStrideElems) {
  u32x4 g0; i32x8 g1;
  tdm_make_desc(g0, g1, lds_addr, gptr, tileW, tileH, rowStrideElems);
  i32x4 gz = {0, 0, 0, 0};
#if defined(__clang_major__) && (__clang_major__ >= 23)
  i32x8 gz8 = {0, 0, 0, 0, 0, 0, 0, 0};
  __builtin_amdgcn_tensor_load_to_lds(g0, g1, gz, gz, gz8, 0);
#else
  __builtin_amdgcn_tensor_load_to_lds(g0, g1, gz, gz, 0);
#endif
}

template <int USE_TDM>
__global__ __launch_bounds__(VPB)
void ks_sim_kernel(const float* __restrict__ emb,
                   const float* __restrict__ noise,
                   float* __restrict__ raw,     // USE_TDM: [N][128] ws; else: [128][N] out
                   float* __restrict__ maxv,    // [128] per-voice max|y|
                   int N) {
  __shared__ float buf[MAXD * VPB];            // delay lines, [k][lane]
  __shared__ float stage[2][CHUNK * VPB];      // double-buffered output tiles

  const int t = threadIdx.x;
  const int v = blockIdx.x * VPB + t;

  // Noise staging aliases the output stage buffers (disjoint in time):
  // a 32x134 f32 tile = 17152 B <= 32768 B of stage space.
  float* nz = &stage[0][0];
  if (USE_TDM) {
    // One DMA for this block's noise slice; overlaps with param math below.
    tdm_load_tile(lds_byte_addr(nz), noise + (size_t)blockIdx.x * VPB * MAXD,
                  /*tileW=*/MAXD, /*tileH=*/VPB, /*rowStride=*/MAXD);
  }

  // ---- per-voice parameters (match reference; e is NOT clipped) ----
  const float e0 = (emb[v * 16 + 0] + 1.0f) * 0.5f;
  const float e1 = (emb[v * 16 + 1] + 1.0f) * 0.5f;
  const float e2 = (emb[v * 16 + 2] + 1.0f) * 0.5f;
  const float excite = (emb[v * 16 + 3] + 1.0f) * 0.5f;
  const float f0 = 120.0f + (880.0f - 120.0f) * e0;
  const float decay_s = 0.06f + (0.6f - 0.06f) * e1;
  const float brightness = 0.2f + (0.98f - 0.2f) * e2;
  // delay can exceed MAXD (f0 < 40 is reachable): keep full value, emulate
  // JAX gather-clamp / scatter-drop against the 134-wide buffer.
  const int delay = (int)fmaxf(2.0f, rintf((float)SR_I / fmaxf(40.0f, f0)));
  int burst_len = delay >> 4; if (burst_len < 4) burst_len = 4;
  const float tau = fmaxf(0.01f, decay_s);
  const float dc  = expf(-1.0f / (tau * (float)SR_I));
  const float b   = fminf(fmaxf(brightness, 0.0f), 0.999f);

  if (USE_TDM) __builtin_amdgcn_s_wait_tensorcnt(0);   // noise tile landed

  // ---- excitation buffer (lane-private column, no barriers needed) ----
  float m = 0.0f;
  for (int k = 0; k < MAXD; ++k) {
    float nv = USE_TDM ? nz[t * MAXD + k] : noise[v * MAXD + k];
    float burst = (k < burst_len) ? 1.0f : 0.0f;
    float ex = excite * burst + (1.0f - excite) * nv;
    float exm = (k < delay) ? ex : 0.0f;
    m = fmaxf(m, fabsf(exm));
    buf[k * VPB + t] = exm;
  }
  const float den = m + 1e-6f;
  for (int k = 0; k < MAXD; ++k) buf[k * VPB + t] = buf[k * VPB + t] / den;

  // ---- sequential string loop, software-pipelined LDS prefetch ----
  float y = 0.0f, mo = 0.0f;
  int idx = 0;
  float x = buf[0 * VPB + t];                  // prefetch step 0 (idx=0 < 134)
  int n = 0, sbuf = 0, chunkId = 0;
  while (n < N) {
    int cur = N - n; if (cur > CHUNK) cur = CHUNK;
    if (USE_TDM && chunkId >= 2) {
      // ensure the TDM that consumed this stage two chunks ago has completed
      __builtin_amdgcn_s_wait_tensorcnt(1);
    }
    float* stg = &stage[sbuf][0];
    for (int c = 0; c < cur; ++c) {
      int idxn = idx + 1; if (idxn >= delay) idxn = 0;
      int rn = (idxn < (MAXD - 1)) ? idxn : (MAXD - 1);   // gather clamp
      float xn = buf[rn * VPB + t];                        // prefetch next x
      float avg = 0.5f * (x + y);
      y = b * y + (1.0f - b) * avg;
      float w = y * dc;
      if (idx < MAXD) {                                    // scatter drop if OOB
        buf[idx * VPB + t] = w;
        if (rn == idx) xn = w;                             // forward past prefetch
      }
      mo = fmaxf(mo, fabsf(y));
      if (USE_TDM) stg[c * VPB + t] = y;                   // conflict-free tile write
      else         raw[(size_t)v * N + n + c] = y;         // fallback direct store
      x = xn; idx = idxn;
    }
    if (USE_TDM) {
      // LDS tile writes must land before the TDM engine reads them
      asm volatile("s_wait_dscnt 0x0" ::: "memory");
      tdm_store_tile(lds_byte_addr(stg),
                     raw + ((size_t)n * NVOICE + (size_t)blockIdx.x * VPB),
                     /*tileW=*/VPB, /*tileH=*/cur, /*rowStride=*/NVOICE);
      sbuf ^= 1;
    }
    n += cur; ++chunkId;
  }
  if (USE_TDM) __builtin_amdgcn_s_wait_tensorcnt(0);
  maxv[v] = mo;
}

// Transpose ws [N][128] -> out [128][N] with per-voice 0.9/(max+1e-6) scaling.
__global__ __launch_bounds__(256)
void ks_norm_transpose(const float* __restrict__ raw,
                       const float* __restrict__ maxv,
                       float* __restrict__ out, int N) {
  __shared__ float tile[32][33];
  const int tx = threadIdx.x, ty = threadIdx.y;
  const int n0 = blockIdx.x * 32, v0 = blockIdx.y * 32;
  for (int j = ty; j < 32; j += 8) {
    int nn = n0 + j;
    tile[j][tx] = (nn < N) ? raw[(size_t)nn * NVOICE + v0 + tx] : 0.0f;
  }
  __syncthreads();
  for (int j = ty; j < 32; j += 8) {
    int vv = v0 + j;
    float s = 0.9f / (maxv[vv] + 1e-6f);
    int nn = n0 + tx;
    if (nn < N) out[(size_t)vv * N + nn] = tile[tx][j] * s;
  }
}

// Fallback: in-place normalize when workspace is too small for the raw buffer.
__global__ void ks_norm_inplace(float* __restrict__ out,
                                const float* __restrict__ maxv,
                                int N, int total) {
  int i = blockIdx.x * blockDim.x + threadIdx.x;
  if (i < total) {
    int vv = i / N;
    out[i] *= 0.9f / (maxv[vv] + 1e-6f);
  }
}

extern "C" void kernel_launch(void* const* d_in, const int* in_sizes, int n_in,
                              void* d_out, int out_size, void* d_ws, size_t ws_size,
                              hipStream_t stream) {
  (void)in_sizes; (void)n_in;
  const float* emb   = (const float*)d_in[0];
  const float* noise = (const float*)d_in[1];
  // d_in[2] = seconds; N is recovered from out_size (= 128 * N).
  float* out = (float*)d_out;
  const int N = out_size / NVOICE;
  const size_t rawBytes = (size_t)N * NVOICE * sizeof(float);
  const int use_tdm = (ws_size >= rawBytes + NVOICE * sizeof(float)) ? 1 : 0;

  if (use_tdm) {
    float* raw  = (float*)d_ws;
    float* maxv = (float*)((char*)d_ws + rawBytes);
    ks_sim_kernel<1><<<NVOICE / VPB, VPB, 0, stream>>>(emb, noise, raw, maxv, N);
    dim3 grid((N + 31) / 32, NVOICE / 32);
    ks_norm_transpose<<<grid, dim3(32, 8), 0, stream>>>(raw, maxv, out, N);
  } else {
    float* maxv = (float*)d_ws;
    ks_sim_kernel<0><<<NVOICE / VPB, VPB, 0, stream>>>(emb, noise, out, maxv, N);
    ks_norm_inplace<<<(out_size + 255) / 256, 256, 0, stream>>>(out, maxv, N, out_size);
  }
}